// PolygonSegmenterGCN_60687887892626
// MI455X (gfx1250) — compile-verified
//
#include <hip/hip_runtime.h>
#include <hip/hip_bf16.h>

#define N_NODES 100000
#define N_EDGES 3200000
#define N_LABEL 1000000

typedef float v2f __attribute__((ext_vector_type(2)));
typedef float v8f __attribute__((ext_vector_type(8)));

// ------------------------------------------------------------- init deg/cnt
__global__ void __launch_bounds__(256) k_init(float* __restrict__ deg,
                                              int* __restrict__ cnt) {
  int i = blockIdx.x * 256 + threadIdx.x;
  if (i < N_NODES) { deg[i] = 1.0f; cnt[i] = 0; }   // self-loop weight = 1
}

// weighted degree + in-degree histogram (one-time pass)
__global__ void __launch_bounds__(256) k_prep(const long long* __restrict__ ei,
                                              const float* __restrict__ w,
                                              float* __restrict__ deg,
                                              int* __restrict__ cnt) {
  int e = blockIdx.x * 256 + threadIdx.x;
  if (e < N_EDGES) {
    int d = (int)ei[N_EDGES + e];
    unsafeAtomicAdd(&deg[d], w[e]);
    atomicAdd(&cnt[d], 1);
  }
}

__global__ void __launch_bounds__(256) k_dinv(float* __restrict__ deg) {
  int i = blockIdx.x * 256 + threadIdx.x;
  if (i < N_NODES) {
    float d = deg[i];
    deg[i] = d > 0.f ? rsqrtf(d) : 0.f;
  }
}

// single-block inclusive scan -> row_ptr (inclusive at i+1) + cursor (exclusive)
__global__ void __launch_bounds__(1024) k_scan(const int* __restrict__ cnt,
                                               int* __restrict__ row_ptr,
                                               int* __restrict__ cursor) {
  __shared__ int sm[1024];
  __shared__ int carry_s;
  int t = threadIdx.x;
  if (t == 0) { carry_s = 0; row_ptr[0] = 0; }
  __syncthreads();
  for (int base = 0; base < N_NODES; base += 1024) {
    int i = base + t;
    int v = (i < N_NODES) ? cnt[i] : 0;
    sm[t] = v;
    __syncthreads();
    for (int off = 1; off < 1024; off <<= 1) {
      int x = (t >= off) ? sm[t - off] : 0;
      __syncthreads();
      sm[t] += x;
      __syncthreads();
    }
    int carry = carry_s;
    if (i < N_NODES) {
      row_ptr[i + 1] = carry + sm[t];
      cursor[i] = carry + sm[t] - v;   // exclusive prefix
    }
    __syncthreads();
    if (t == 1023) carry_s = carry + sm[1023];
    __syncthreads();
  }
}

// CSR fill: csr_src / fused symmetric norm, grouped by destination node
__global__ void __launch_bounds__(256) k_fill(const long long* __restrict__ ei,
                                              const float* __restrict__ w,
                                              const float* __restrict__ dinv,
                                              int* __restrict__ cursor,
                                              int* __restrict__ csr_src,
                                              float* __restrict__ csr_nrm) {
  int e = blockIdx.x * 256 + threadIdx.x;
  if (e < N_EDGES) {
    int s = (int)ei[e];
    int d = (int)ei[N_EDGES + e];
    int pos = atomicAdd(&cursor[d], 1);
    csr_src[pos] = s;
    csr_nrm[pos] = dinv[s] * w[e] * dinv[d];
  }
}

// ---------------------------------------------------------------- WMMA GEMM
// out[N_NODES x 64] = act(A[N_NODES x K]) @ W[K x 64], fp32 v_wmma_f32_16x16x4_f32
template <int K, bool RELU>
__global__ void __launch_bounds__(256) k_gemm_wmma(const float* __restrict__ A,
                                                   const float* __restrict__ W,
                                                   float* __restrict__ out) {
  int wave = threadIdx.x >> 5;
  int lane = threadIdx.x & 31;
  int tile = blockIdx.x * 8 + wave;   // (N_NODES/16)*4 tiles, exact fit
  int tm = tile >> 2;
  int tn = tile & 3;
  int half = lane >> 4;               // A 16x4 f32: lanes 0-15 -> K{0,1}; 16-31 -> K{2,3}
  int l = lane & 15;
  int row = tm * 16 + l;
  int col = tn * 16 + l;

  v8f c = {0.f, 0.f, 0.f, 0.f, 0.f, 0.f, 0.f, 0.f};
  const float* arow = A + (size_t)row * K;
#pragma unroll
  for (int kk = 0; kk < K; kk += 4) {
    float a0 = arow[kk + 2 * half + 0];
    float a1 = arow[kk + 2 * half + 1];
    if (RELU) { a0 = fmaxf(a0, 0.f); a1 = fmaxf(a1, 0.f); }
    v2f a; a.x = a0; a.y = a1;
    v2f b;
    b.x = W[(kk + 2 * half + 0) * 64 + col];
    b.y = W[(kk + 2 * half + 1) * 64 + col];
    c = __builtin_amdgcn_wmma_f32_16x16x4_f32(false, a, false, b, (short)0, c,
                                              false, false);
  }
  float* orow = out + ((size_t)tm * 16 + half * 8) * 64 + col;
#pragma unroll
  for (int v = 0; v < 8; ++v) orow[(size_t)v * 64] = c[v];
}

// --------------------------------------------------- pull aggregation (no atomics)
// acc[n] = bias + dinv[n]^2*hw[n] + sum_{e in CSR[n]} nrm[e]*hw[src[e]]
// one wave32 per node; lane owns channels {2l, 2l+1}; edge metadata read
// coalesced then broadcast via readlane -> uniform-base b64 gathers.
__global__ void __launch_bounds__(256) k_aggregate(float* __restrict__ acc,
                                                   const float* __restrict__ hw,
                                                   const int* __restrict__ row_ptr,
                                                   const int* __restrict__ csr_src,
                                                   const float* __restrict__ csr_nrm,
                                                   const float* __restrict__ bias,
                                                   const float* __restrict__ dinv) {
  int wave = threadIdx.x >> 5;
  int lane = threadIdx.x & 31;
  int n = blockIdx.x * 8 + wave;
  if (n >= N_NODES) return;

  float di = dinv[n];
  v2f hself = ((const v2f*)(hw + (size_t)n * 64))[lane];
  float a0 = bias[2 * lane]     + di * di * hself.x;
  float a1 = bias[2 * lane + 1] + di * di * hself.y;

  int beg = row_ptr[n], end = row_ptr[n + 1];
  for (int base = beg; base < end; base += 32) {
    int i = base + lane;
    int s = 0;
    float wv = 0.f;
    if (i < end) { s = csr_src[i]; wv = csr_nrm[i]; }
    int m = end - base; if (m > 32) m = 32;
    for (int j = 0; j < m; ++j) {
      int sj = __builtin_amdgcn_readlane(s, j);                       // SGPR base
      float wj = __int_as_float(__builtin_amdgcn_readlane(__float_as_int(wv), j));
      v2f h = ((const v2f*)(hw + (size_t)sj * 64))[lane];
      a0 += wj * h.x;
      a1 += wj * h.y;
    }
  }
  v2f r; r.x = a0; r.y = a1;
  ((v2f*)(acc + (size_t)n * 64))[lane] = r;
}

// ------------------------------------------------------------------- decode
__global__ void __launch_bounds__(256) k_decode(const float* __restrict__ z,
                                                const long long* __restrict__ eli,
                                                float* __restrict__ out) {
  int wave = threadIdx.x >> 5;
  int lane = threadIdx.x & 31;
  int e = blockIdx.x * 8 + wave;
  if (e >= N_LABEL) return;
  int a = (int)eli[e];
  int b = (int)eli[N_LABEL + e];
  v2f xa = ((const v2f*)(z + (size_t)a * 64))[lane];
  v2f xb = ((const v2f*)(z + (size_t)b * 64))[lane];
  float p = xa.x * xb.x + xa.y * xb.y;
#pragma unroll
  for (int off = 16; off > 0; off >>= 1) p += __shfl_xor(p, off, 32);
  if (lane == 0) out[e] = p;
}

// ---------------------------------------------------------------- launcher
extern "C" void kernel_launch(void* const* d_in, const int* in_sizes, int n_in,
                              void* d_out, int out_size, void* d_ws, size_t ws_size,
                              hipStream_t stream) {
  const float* x       = (const float*)d_in[0];
  const float* ew      = (const float*)d_in[1];
  const long long* ei  = (const long long*)d_in[2];
  const long long* eli = (const long long*)d_in[3];
  const float* Wt[5] = {(const float*)d_in[4], (const float*)d_in[6],
                        (const float*)d_in[8], (const float*)d_in[10],
                        (const float*)d_in[12]};
  const float* bs[5] = {(const float*)d_in[5], (const float*)d_in[7],
                        (const float*)d_in[9], (const float*)d_in[11],
                        (const float*)d_in[13]};
  float* outp = (float*)d_out;

  char* ws = (char*)d_ws;
  float* deg     = (float*)ws; ws += sizeof(float) * N_NODES;      // becomes dinv
  int*   cnt     = (int*)ws;   ws += sizeof(int) * N_NODES;
  int*   row_ptr = (int*)ws;   ws += sizeof(int) * (N_NODES + 4);  // N+1, padded
  int*   cursor  = (int*)ws;   ws += sizeof(int) * N_NODES;
  int*   csr_src = (int*)ws;   ws += sizeof(int) * N_EDGES;
  float* csr_nrm = (float*)ws; ws += sizeof(float) * N_EDGES;
  float* hw      = (float*)ws; ws += sizeof(float) * (size_t)N_NODES * 64;
  float* acc     = (float*)ws; ws += sizeof(float) * (size_t)N_NODES * 64;

  const dim3 B(256);
  const int gN = (N_NODES + 255) / 256;
  const int gE = (N_EDGES + 255) / 256;
  const int gWave = (N_NODES + 7) / 8;     // wave-per-node kernels
  const int gGemm = (N_NODES / 16) * 4 / 8;
  const int gDec  = (N_LABEL + 7) / 8;

  k_init<<<gN, B, 0, stream>>>(deg, cnt);
  k_prep<<<gE, B, 0, stream>>>(ei, ew, deg, cnt);
  k_dinv<<<gN, B, 0, stream>>>(deg);
  k_scan<<<1, 1024, 0, stream>>>(cnt, row_ptr, cursor);
  k_fill<<<gE, B, 0, stream>>>(ei, ew, deg, cursor, csr_src, csr_nrm);

  // layer 1: input x (K=32), no input activation
  k_gemm_wmma<32, false><<<gGemm, B, 0, stream>>>(x, Wt[0], hw);
  k_aggregate<<<gWave, B, 0, stream>>>(acc, hw, row_ptr, csr_src, csr_nrm,
                                       bs[0], deg);
  // layers 2..5: input relu(acc); layer-5 output (z) stays un-activated in acc
  for (int layer = 1; layer < 5; ++layer) {
    k_gemm_wmma<64, true><<<gGemm, B, 0, stream>>>(acc, Wt[layer], hw);
    k_aggregate<<<gWave, B, 0, stream>>>(acc, hw, row_ptr, csr_src, csr_nrm,
                                         bs[layer], deg);
  }

  k_decode<<<gDec, B, 0, stream>>>(acc, eli, outp);
}